// TensorProductExpansion_49735721287938
// MI455X (gfx1250) — compile-verified
//
#include <hip/hip_runtime.h>

// TensorProductExpansion: out[c,i0,i1,i2,i3] = prod_n (tensor+bias)[c,n,i_n]
// C=2048, N=4, L=16 -> 512 MB f32 output; pure store-bandwidth kernel.

typedef __attribute__((ext_vector_type(2))) float v2f;
typedef __attribute__((ext_vector_type(8))) float v8f;

__global__ __launch_bounds__(256) void tpexp_kernel(const float* __restrict__ tensor,
                                                    const float* __restrict__ bias,
                                                    float* __restrict__ out)
{
    const int ch   = blockIdx.x;      // one channel per block
    const int t    = threadIdx.x;     // 256 threads = 8 wave32
    const int lane = t & 31;
    const int wave = t >> 5;

    const float* tch = tensor + (size_t)ch * 64;   // 4 rows x 16
    const float* bch = bias   + (size_t)ch * 64;

    // ---------- p01 = outer(a, b) via V_WMMA_F32_16X16X4_F32 (exact f32) ----------
    // A (16x4 f32): lanes 0-15 hold K=0 (VGPR0) and K=1 (VGPR1); lanes 16-31 hold K=2,3.
    // Put vector a in column K=0, everything else 0 -> D[M][N] = a[M]*b[N].
    float aval = 0.0f, bval = 0.0f;
    if (lane < 16) {
        aval = tch[lane]      + bch[lane];        // row 0 of tb -> a
        bval = tch[16 + lane] + bch[16 + lane];   // row 1 of tb -> b
    }
    v2f A; A.x = aval; A.y = 0.0f;
    v2f B; B.x = bval; B.y = 0.0f;
    v8f Cz = {};
    // 8 args: (neg_a, A, neg_b, B, c_mod, C, reuse_a, reuse_b)
    v8f D = __builtin_amdgcn_wmma_f32_16x16x4_f32(false, A, false, B, (short)0, Cz,
                                                  false, false);

    // Spill D to a per-wave LDS copy: D layout -> lanes 0-15: (M=r, N=lane),
    // lanes 16-31: (M=8+r, N=lane-16).  p01[M*16+N] == p01[i0*16+i1].
    __shared__ float p01[8][256];
    {
        const int mbase = (lane >> 4) * 8;
        const int n     = lane & 15;
#pragma unroll
        for (int r = 0; r < 8; ++r)
            p01[wave][(mbase + r) * 16 + n] = D[r];
    }
    __syncthreads();

    // ---------- expansion: out[ch, i, j] = p01[i] * c[j>>4] * d[j&15] ----------
    // Thread t owns j = (t&63)*4 .. +3 (fixed) and iterates i = (t>>6)*64 .. +63.
    // Within a wave, lanes write consecutive float4 -> each store = 512B contiguous.
    const int tl = t & 63;
    const int ib = (t >> 6) * 64;
    const int jh = tl >> 2;          // j>>4 : index into row c
    const int dl = (tl & 3) * 4;     // j&15 : base index into row d

    const float cv  = tch[32 + jh] + bch[32 + jh];
    const float cd0 = cv * (tch[48 + dl + 0] + bch[48 + dl + 0]);
    const float cd1 = cv * (tch[48 + dl + 1] + bch[48 + dl + 1]);
    const float cd2 = cv * (tch[48 + dl + 2] + bch[48 + dl + 2]);
    const float cd3 = cv * (tch[48 + dl + 3] + bch[48 + dl + 3]);

    float* __restrict__ op = out + (size_t)ch * 65536 + (size_t)ib * 256 + (size_t)tl * 4;
    const float* sp = &p01[wave][ib];   // i is wave-uniform each iteration -> ds broadcast

#pragma unroll
    for (int k = 0; k < 64; ++k) {
        const float s = sp[k];
        float4 v = make_float4(s * cd0, s * cd1, s * cd2, s * cd3);
        *reinterpret_cast<float4*>(op + (size_t)k * 256) = v;   // global_store_b128
    }
}

extern "C" void kernel_launch(void* const* d_in, const int* in_sizes, int n_in,
                              void* d_out, int out_size, void* d_ws, size_t ws_size,
                              hipStream_t stream)
{
    const float* tensor = (const float*)d_in[0];
    const float* bias   = (const float*)d_in[1];
    float* out = (float*)d_out;

    const int C = in_sizes[0] / 64;   // (C,4,16) f32
    tpexp_kernel<<<C, 256, 0, stream>>>(tensor, bias, out);
}